// TemporalGNN_46986942218820
// MI455X (gfx1250) — compile-verified
//
#include <hip/hip_runtime.h>
#include <hip/hip_bf16.h>

// ---------------- problem constants (from reference) ----------------
#define NN 50000        // num nodes
#define RR 50           // num relations
#define BB 30           // num bases
#define H1 64           // layer-1 width
#define H2 32           // layer-2 width
#define EE 1600000      // num edges
#define NC1 (NN * H1)   // 3,200,000 columns of weight1 (= N*H1)
#define W2C (H1 * H2)   // 2048 columns of weight2

typedef __attribute__((ext_vector_type(2))) float v2f;
typedef __attribute__((ext_vector_type(8))) float v8f;

// f32 WMMA 16x16x4: D = A(16x4) * B(4x16) + C(16x16), exact f32 FMA chain.
__device__ __forceinline__ v8f wmma_f32(v2f a, v2f b, v8f c) {
    return __builtin_amdgcn_wmma_f32_16x16x4_f32(
        /*neg_a=*/false, a, /*neg_b=*/false, b,
        /*c_mod=*/(short)0, c, /*reuse_a=*/false, /*reuse_b=*/false);
}

// ---------------- utility: zero a float buffer (grid-stride) ----------------
__global__ void zero_f32(float* __restrict__ p, long n) {
    long i = (long)blockIdx.x * blockDim.x + threadIdx.x;
    long stride = (long)gridDim.x * blockDim.x;
    for (; i < n; i += stride) p[i] = 0.0f;
}

// ---------------- K1: per-(dst,rel) in-degree -------------------------------
__global__ void deg_kernel(const int* __restrict__ dst,
                           const int* __restrict__ etype,
                           float* __restrict__ deg) {
    long e = (long)blockIdx.x * blockDim.x + threadIdx.x;
    if (e >= EE) return;
    atomicAdd(&deg[(long)dst[e] * RR + etype[e]], 1.0f);
}

// ---------------- K2: weight1 = comp1[50x30] @ basis1[30 x 3.2M] ------------
// One wave per 16-column tile; 4 M-tiles (M padded 50->64), 8 K-steps (30->32).
// A layout (16x4 f32): lanes 0-15 hold K=+0/+1 in vgpr0/1, lanes 16-31 K=+2/+3.
// B layout (4x16):     vgpr i: low lanes K=i, high lanes K=i+2, N=lane&15.
// C layout (16x16):    vgpr v: low lanes M=v, high lanes M=v+8, N=lane&15.
__global__ void __launch_bounds__(256) w1_gemm(const float* __restrict__ comp1,
                                               const float* __restrict__ basis1,
                                               float* __restrict__ w1) {
    __shared__ float sc[64 * 32];   // zero-padded comp1 [row 0..63][k 0..31]
    for (int i = threadIdx.x; i < 64 * 32; i += blockDim.x) {
        int row = i >> 5, k = i & 31;
        sc[i] = (row < RR && k < BB) ? comp1[row * BB + k] : 0.0f;
    }
    __syncthreads();

    const int wave = (blockIdx.x * blockDim.x + threadIdx.x) >> 5;   // 200,000 waves exactly
    const int lane = threadIdx.x & 31;
    const int half = lane >> 4;      // 0: lanes 0-15, 1: lanes 16-31
    const int l    = lane & 15;
    const long col = (long)wave * 16 + l;

    v8f acc[4] = {v8f{}, v8f{}, v8f{}, v8f{}};

    for (int ks = 0; ks < 8; ++ks) {
        v2f bf;
#pragma unroll
        for (int i = 0; i < 2; ++i) {
            int k = ks * 4 + i + 2 * half;
            bf[i] = (k < BB) ? basis1[(long)k * NC1 + col] : 0.0f;
        }
#pragma unroll
        for (int mt = 0; mt < 4; ++mt) {
            v2f af;
#pragma unroll
            for (int i = 0; i < 2; ++i) {
                int k = ks * 4 + i + 2 * half;
                af[i] = sc[(mt * 16 + l) * 32 + k];
            }
            acc[mt] = wmma_f32(af, bf, acc[mt]);
        }
    }

#pragma unroll
    for (int mt = 0; mt < 4; ++mt) {
#pragma unroll
        for (int v = 0; v < 8; ++v) {
            int row = mt * 16 + v + 8 * half;
            if (row < RR) w1[(long)row * NC1 + col] = acc[mt][v];
        }
    }
}

// ---------------- K3: layer-1 edge scatter ----------------------------------
// One wave per edge; each lane handles 2 of the 64 features (coalesced 256B row).
__global__ void gather1(const int* __restrict__ src, const int* __restrict__ dst,
                        const int* __restrict__ etype, const float* __restrict__ deg,
                        const float* __restrict__ w1, float* __restrict__ hacc) {
    long t = (long)blockIdx.x * blockDim.x + threadIdx.x;
    long e = t >> 5;
    if (e >= EE) return;
    int j = (int)(t & 31) * 2;
    int s = src[e], d = dst[e], r = etype[e];
    float sc = 1.0f / fmaxf(deg[(long)d * RR + r], 1.0f);
    const float* wrow = w1 + (long)r * NC1 + (long)s * H1 + j;
    atomicAdd(&hacc[(long)d * H1 + j],     wrow[0] * sc);
    atomicAdd(&hacc[(long)d * H1 + j + 1], wrow[1] * sc);
}

// ---------------- K4: h = relu(agg1 + root1 + bias1) ------------------------
__global__ void h_final(float* __restrict__ hacc, const float* __restrict__ root1,
                        const float* __restrict__ bias1) {
    long i = (long)blockIdx.x * blockDim.x + threadIdx.x;
    if (i >= (long)NN * H1) return;
    float v = hacc[i] + root1[i] + bias1[i & (H1 - 1)];
    hacc[i] = v > 0.0f ? v : 0.0f;
}

// ---------------- K5: weight2 = comp2[50x30] @ basis2[30x2048] (tiny) -------
__global__ void w2_kernel(const float* __restrict__ comp2,
                          const float* __restrict__ basis2,
                          float* __restrict__ w2) {
    int i = blockIdx.x * blockDim.x + threadIdx.x;
    if (i >= RR * W2C) return;
    int r = i / W2C, c = i % W2C;
    float s = 0.0f;
#pragma unroll
    for (int b = 0; b < BB; ++b) s = fmaf(comp2[r * BB + b], basis2[b * W2C + c], s);
    w2[i] = s;
}

// ---------------- K6: x_all[n,r,:] = h[n,:] @ weight2[r] (WMMA) -------------
// One wave per (node-tile, relation): 16 k-steps (K=64), 2 column tiles (H2=32).
__global__ void __launch_bounds__(256) xall_gemm(const float* __restrict__ h,
                                                 const float* __restrict__ w2,
                                                 float* __restrict__ xall) {
    const int wave = (blockIdx.x * blockDim.x + threadIdx.x) >> 5;
    const int NTILES = NN / 16;                 // 3125 exactly
    if (wave >= NTILES * RR) return;            // wave-uniform exit (EXEC stays full)
    const int lane = threadIdx.x & 31;
    const int half = lane >> 4;
    const int l    = lane & 15;
    const int r    = wave % RR;
    const int node0 = (wave / RR) * 16;
    const float* w2r = w2 + r * W2C;

    v8f acc[2] = {v8f{}, v8f{}};
    for (int ks = 0; ks < 16; ++ks) {
        v2f af;
#pragma unroll
        for (int i = 0; i < 2; ++i) {
            int k = ks * 4 + i + 2 * half;
            af[i] = h[(long)(node0 + l) * H1 + k];
        }
#pragma unroll
        for (int jt = 0; jt < 2; ++jt) {
            v2f bf;
#pragma unroll
            for (int i = 0; i < 2; ++i) {
                int k = ks * 4 + i + 2 * half;
                bf[i] = w2r[k * H2 + jt * 16 + l];
            }
            acc[jt] = wmma_f32(af, bf, acc[jt]);
        }
    }
#pragma unroll
    for (int jt = 0; jt < 2; ++jt) {
#pragma unroll
        for (int v = 0; v < 8; ++v) {
            int node = node0 + v + 8 * half;
            xall[((long)node * RR + r) * H2 + jt * 16 + l] = acc[jt][v];
        }
    }
}

// ---------------- K7: layer-2 edge scatter ----------------------------------
// 16 threads per edge, 2 features each (coalesced 128B row).
__global__ void gather2(const int* __restrict__ src, const int* __restrict__ dst,
                        const int* __restrict__ etype, const float* __restrict__ deg,
                        const float* __restrict__ xall, float* __restrict__ out) {
    long t = (long)blockIdx.x * blockDim.x + threadIdx.x;
    long e = t >> 4;
    if (e >= EE) return;
    int j = (int)(t & 15) * 2;
    int s = src[e], d = dst[e], r = etype[e];
    float sc = 1.0f / fmaxf(deg[(long)d * RR + r], 1.0f);
    const float* xr = xall + ((long)s * RR + r) * H2 + j;
    atomicAdd(&out[(long)d * H2 + j],     xr[0] * sc);
    atomicAdd(&out[(long)d * H2 + j + 1], xr[1] * sc);
}

// ---------------- K8: out += h @ root2 + bias2 (WMMA, fused finalize) -------
__global__ void __launch_bounds__(256) out_final(const float* __restrict__ h,
                                                 const float* __restrict__ root2,
                                                 const float* __restrict__ bias2,
                                                 float* __restrict__ out) {
    const int wave = (blockIdx.x * blockDim.x + threadIdx.x) >> 5;
    const int NTILES = NN / 16;
    if (wave >= NTILES) return;                 // wave-uniform exit
    const int lane = threadIdx.x & 31;
    const int half = lane >> 4;
    const int l    = lane & 15;
    const int node0 = wave * 16;

    v8f acc[2] = {v8f{}, v8f{}};
    for (int ks = 0; ks < 16; ++ks) {
        v2f af;
#pragma unroll
        for (int i = 0; i < 2; ++i) {
            int k = ks * 4 + i + 2 * half;
            af[i] = h[(long)(node0 + l) * H1 + k];
        }
#pragma unroll
        for (int jt = 0; jt < 2; ++jt) {
            v2f bf;
#pragma unroll
            for (int i = 0; i < 2; ++i) {
                int k = ks * 4 + i + 2 * half;
                bf[i] = root2[k * H2 + jt * 16 + l];
            }
            acc[jt] = wmma_f32(af, bf, acc[jt]);
        }
    }
#pragma unroll
    for (int jt = 0; jt < 2; ++jt) {
        int j = jt * 16 + l;
        float b = bias2[j];
#pragma unroll
        for (int v = 0; v < 8; ++v) {
            int node = node0 + v + 8 * half;
            long idx = (long)node * H2 + j;
            out[idx] = out[idx] + acc[jt][v] + b;   // each element owned by exactly one lane
        }
    }
}

// ---------------- driver ----------------------------------------------------
extern "C" void kernel_launch(void* const* d_in, const int* in_sizes, int n_in,
                              void* d_out, int out_size, void* d_ws, size_t ws_size,
                              hipStream_t stream) {
    const int*   edge_index = (const int*)d_in[0];     // [2, E]
    const int*   etype      = (const int*)d_in[1];     // [E]
    /* d_in[2] edge_timestamps: unused by reference */
    const float* basis1 = (const float*)d_in[3];       // [B, N, H1]
    const float* comp1  = (const float*)d_in[4];       // [R, B]
    const float* root1  = (const float*)d_in[5];       // [N, H1]
    const float* bias1  = (const float*)d_in[6];       // [H1]
    const float* basis2 = (const float*)d_in[7];       // [B, H1, H2]
    const float* comp2  = (const float*)d_in[8];       // [R, B]
    const float* root2  = (const float*)d_in[9];       // [H1, H2]
    const float* bias2  = (const float*)d_in[10];      // [H2]
    float* out = (float*)d_out;                        // [N, H2]

    // workspace layout (floats): deg | hacc | w2 | big (weight1, reused as x_all)
    float* ws   = (float*)d_ws;
    float* deg  = ws;                                  // N*R   = 2,500,000
    float* hacc = deg  + (long)NN * RR;                // N*H1  = 3,200,000
    float* w2   = hacc + (long)NN * H1;                // R*2048 = 102,400
    float* big  = w2   + (long)RR * W2C;               // 160,000,000 (w1) / 80,000,000 (x_all)

    const int* src = edge_index;
    const int* dst = edge_index + EE;

    // zero accumulators (deg+hacc contiguous) and output
    zero_f32<<<2048, 256, 0, stream>>>(deg, (long)NN * RR + (long)NN * H1);
    zero_f32<<<1024, 256, 0, stream>>>(out, (long)NN * H2);

    // per-(dst,rel) degrees
    deg_kernel<<<(EE + 255) / 256, 256, 0, stream>>>(dst, etype, deg);

    // layer 1: materialize weight1, scatter messages, finalize h
    w1_gemm<<<25000, 256, 0, stream>>>(comp1, basis1, big);          // 200,000 waves
    gather1<<<200000, 256, 0, stream>>>(src, dst, etype, deg, big, hacc);
    h_final<<<((long)NN * H1 + 255) / 256, 256, 0, stream>>>(hacc, root1, bias1);

    // layer 2: weight2, x_all (reuses `big`), scatter, root/bias finalize
    w2_kernel<<<(RR * W2C + 255) / 256, 256, 0, stream>>>(comp2, basis2, w2);
    xall_gemm<<<(3125 * RR * 32 + 255) / 256, 256, 0, stream>>>(hacc, w2, big);
    gather2<<<100000, 256, 0, stream>>>(src, dst, etype, deg, big, out);
    out_final<<<(3125 * 32 + 255) / 256, 256, 0, stream>>>(hacc, root2, bias2, out);
}